// CBMAELoss_70274254897863
// MI455X (gfx1250) — compile-verified
//
#include <hip/hip_runtime.h>
#include <hip/hip_bf16.h>

typedef __attribute__((ext_vector_type(2))) float v2f;
typedef __attribute__((ext_vector_type(8))) float v8f;

#define B_DIM   16
#define C_DIM   12
#define HW      384
#define LK      96
#define PLANE   (HW * HW)      /* 147456 */
#define LKPLANE (LK * LK)      /* 9216   */
#define XG      (HW / 4)       /* 96 x-groups of 4 pixels */
#define NTHREADS (B_DIM * HW * XG)   /* 589824 */
#define BLOCK   256
#define NBLOCKS (NTHREADS / BLOCK)   /* 2304   */
#define NPART   (NTHREADS / 16)      /* 36864 partials: 2 per wave32 */

__device__ __forceinline__ float wtrue_f(float v) {
    // Thresholds exactly as Python computes them (double) then rounded to f32.
    const float T0 = (float)((31.0  - 33.44) / 47.54);
    const float T1 = (float)((74.0  - 33.44) / 47.54);
    const float T2 = (float)((133.0 - 33.44) / 47.54);
    const float T3 = (float)((181.0 - 33.44) / 47.54);
    const float T4 = (float)((255.0 - 33.44) / 47.54);
    // Sequential (cascading) replace chain, faithful to in-place semantics.
    v = v < T0 ? 1.0f  : v;
    v = v < T1 ? 2.0f  : v;
    v = v < T2 ? 5.0f  : v;
    v = v < T3 ? 10.0f : v;
    v = v < T4 ? 30.0f : v;
    return v;
}

__global__ __launch_bounds__(BLOCK) void cbmae_main_kernel(
    const float* __restrict__ y_pre,
    const float* __restrict__ y_true,
    const float* __restrict__ x_l,
    const float* __restrict__ w_LK,
    float* __restrict__ wave_partials)
{
    const int tid = blockIdx.x * BLOCK + threadIdx.x;
    const int b   = tid / (HW * XG);
    const int rem = tid - b * (HW * XG);
    const int y   = rem / XG;
    const int xg  = rem - y * XG;
    const int x   = xg << 2;

    // Vertical bilinear params: src_y = y/4 - 0.375 (half-pixel, scale 4).
    const int r = y & 3;
    const int k = y >> 2;
    int ylo = k - (r < 2 ? 1 : 0);
    int yhi = ylo + 1;
    ylo = ylo < 0 ? 0 : ylo;
    yhi = yhi > (LK - 1) ? (LK - 1) : yhi;
    const float wy  = (r == 0) ? 0.625f : (r == 1) ? 0.875f : (r == 2) ? 0.125f : 0.375f;
    const float wyl = 1.0f - wy;

    // Horizontal taps: 4 pixels x=4k..4k+3 need columns {k-1, k, k+1} (clamped).
    const int cm1 = xg > 0 ? xg - 1 : 0;
    const int cc0 = xg;
    const int cp1 = xg < (LK - 1) ? xg + 1 : (LK - 1);

    const size_t pix = (size_t)y * HW + x;
    const float4 xl = *(const float4*)(x_l + (size_t)b * PLANE + pix);

    const float* yt_p = y_true + (size_t)b * C_DIM * PLANE + pix;
    const float* yp_p = y_pre  + (size_t)b * C_DIM * PLANE + pix;
    const float* lk_p = w_LK   + (size_t)b * C_DIM * LKPLANE;

    const int tlo = ylo * LK;
    const int thi = yhi * LK;

    // Running exclusive cumsums (registers; linearity lets us cumsum the taps).
    float accA = 0.0f, accB = 0.0f, accC = 0.0f;               // row-blended w_LK taps
    float run0 = 0.0f, run1 = 0.0f, run2 = 0.0f, run3 = 0.0f;  // w_c_v cumsum
    float partial = 0.0f;

    #pragma unroll 1
    for (int c = 0; c < C_DIM; ++c) {
        const float4 yt = *(const float4*)yt_p;
        const float4 yp = *(const float4*)yp_p;
        if (c + 1 < C_DIM) {
            // locality 3 -> WGP-scope prefetch: pull into all cache levels.
            __builtin_prefetch(yt_p + PLANE, 0, 3);   // global_prefetch_b8
            __builtin_prefetch(yp_p + PLANE, 0, 3);
        }

        // Bilinear values for the 4 pixels from accumulated (exclusive) taps.
        const float i0 = 0.375f * accA + 0.625f * accB;
        const float i1 = 0.125f * accA + 0.875f * accB;
        const float i2 = 0.875f * accB + 0.125f * accC;
        const float i3 = 0.625f * accB + 0.375f * accC;

        partial += (wtrue_f(yt.x) + run0 + i0) * fabsf(yp.x - yt.x);
        partial += (wtrue_f(yt.y) + run1 + i1) * fabsf(yp.y - yt.y);
        partial += (wtrue_f(yt.z) + run2 + i2) * fabsf(yp.z - yt.z);
        partial += (wtrue_f(yt.w) + run3 + i3) * fabsf(yp.w - yt.w);

        run0 += fabsf(yt.x - xl.x);
        run1 += fabsf(yt.y - xl.y);
        run2 += fabsf(yt.z - xl.z);
        run3 += fabsf(yt.w - xl.w);

        // Fold this channel's w_LK taps into the exclusive-cumsum accumulators.
        accA += wyl * lk_p[tlo + cm1] + wy * lk_p[thi + cm1];
        accB += wyl * lk_p[tlo + cc0] + wy * lk_p[thi + cc0];
        accC += wyl * lk_p[tlo + cp1] + wy * lk_p[thi + cp1];

        yt_p += PLANE;
        yp_p += PLANE;
        lk_p += LKPLANE;
    }

    // ---- Wave32 reduction via V_WMMA_F32_16X16X4_F32 (full f32 precision) ----
    // A: each lane contributes its partial in A-VGPR0 (one unique slot of the
    // 16x4 A matrix), 0 in A-VGPR1. B: all-ones 4x16 (layout-proof).
    // D[m][n] = row-sum_m. Lane 0 holds D[0..7][0], lane 16 holds D[8..15][0];
    // their two 8-element sums together cover all 32 lane partials, so we store
    // two partials per wave and skip any cross-lane shuffle.
    v2f a;  a.x = partial; a.y = 0.0f;
    v2f bo; bo.x = 1.0f;   bo.y = 1.0f;
    v8f cz = {};
    v8f d = __builtin_amdgcn_wmma_f32_16x16x4_f32(
        /*neg_a=*/false, a, /*neg_b=*/false, bo,
        /*c_mod=*/(short)0, cz, /*reuse_a=*/false, /*reuse_b=*/false);

    const float s = ((d[0] + d[1]) + (d[2] + d[3])) + ((d[4] + d[5]) + (d[6] + d[7]));

    if ((threadIdx.x & 15) == 0) {      // lanes 0 and 16 of each wave
        wave_partials[tid >> 4] = s;    // slot = 2*waveId + (lane>>4)
    }
}

__global__ __launch_bounds__(256) void cbmae_reduce_kernel(
    const float* __restrict__ wave_partials, float* __restrict__ out)
{
    __shared__ double sh[256];
    double acc = 0.0;
    // Fixed per-thread strided order -> deterministic.
    for (int i = threadIdx.x; i < NPART; i += 256)
        acc += (double)wave_partials[i];
    sh[threadIdx.x] = acc;
    __syncthreads();
    for (int off = 128; off > 0; off >>= 1) {
        if (threadIdx.x < off) sh[threadIdx.x] += sh[threadIdx.x + off];
        __syncthreads();
    }
    if (threadIdx.x == 0) {
        const double n = (double)((size_t)B_DIM * C_DIM * PLANE);
        out[0] = (float)(sh[0] / n);
    }
}

extern "C" void kernel_launch(void* const* d_in, const int* in_sizes, int n_in,
                              void* d_out, int out_size, void* d_ws, size_t ws_size,
                              hipStream_t stream) {
    const float* y_pre  = (const float*)d_in[0];
    const float* y_true = (const float*)d_in[1];
    const float* x_l    = (const float*)d_in[2];
    const float* w_LK   = (const float*)d_in[3];
    float* partials = (float*)d_ws;   // NPART floats = 144 KB scratch
    float* out      = (float*)d_out;

    cbmae_main_kernel<<<NBLOCKS, BLOCK, 0, stream>>>(y_pre, y_true, x_l, w_LK, partials);
    cbmae_reduce_kernel<<<1, 256, 0, stream>>>(partials, out);
}